// Agent_69965017252670
// MI455X (gfx1250) — compile-verified
//
#include <hip/hip_runtime.h>
#include <hip/hip_bf16.h>
#include <math.h>

// Problem constants (from reference)
#define BB 8
#define VV 8
#define NN 1000
#define DD 256
#define HH 8
#define KSZ 32
#define FNODE 8
#define NTILES 63   // ceil(1000/16)
#define NPAD 1008   // 63*16

typedef __attribute__((ext_vector_type(2))) float v2f;
typedef __attribute__((ext_vector_type(4))) float v4f;
typedef __attribute__((ext_vector_type(8))) float v8f;
typedef __attribute__((ext_vector_type(4))) int v4i;

#if __has_builtin(__builtin_amdgcn_global_load_async_to_lds_b128)
#define HAVE_ASYNC 1
// async loads retire in order; wait for older tile only (4 ops outstanding)
#define ASYNC_WAIT_ALL()  asm volatile("s_wait_asynccnt 0x0" ::: "memory")
#define ASYNC_WAIT_PREV() asm volatile("s_wait_asynccnt 0x4" ::: "memory")
#else
#define HAVE_ASYNC 0
#define ASYNC_WAIT_ALL()
#define ASYNC_WAIT_PREV()
#endif

// V_WMMA_F32_16X16X4_F32 : D(16x16 f32) = A(16x4 f32) * B(4x16 f32) + C
// A layout: lanes 0-15 hold M=0..15 {K0,K1}, lanes 16-31 hold {K2,K3}
// B layout: lanes 0-15 hold N=0..15 {K0,K1}, lanes 16-31 hold {K2,K3}
// C/D layout: VGPR r, lane L -> (M = r + 8*(L>=16), N = L&15)
__device__ __forceinline__ v8f wmma4(v2f a, v2f b, v8f c) {
  return __builtin_amdgcn_wmma_f32_16x16x4_f32(
      /*neg_a=*/false, a, /*neg_b=*/false, b,
      /*c_mod=*/(short)0, c, /*reuse_a=*/false, /*reuse_b=*/false);
}

// Stage one 16-node x 32-f32 tile (2048 B) of a K/V plane into LDS.
// Per-lane 4 chunks of 16 B; `maxfo` clamps the global float-offset in-bounds
// for the ragged tail tile (clamped chunks land in unused slots).
__device__ __forceinline__ void stage_tile(const float* __restrict__ g,
                                           float* l, int lane, long maxfo) {
#pragma unroll
  for (int i = 0; i < 4; ++i) {
    long fo = (long)(i * 32 + lane) * 4;
    if (fo > maxfo) fo = maxfo;
#if HAVE_ASYNC
    __builtin_amdgcn_global_load_async_to_lds_b128(
        (v4i*)(g + fo), (v4i*)(l + (i * 32 + lane) * 4), 0, 0);
#else
    *(v4f*)(l + (i * 32 + lane) * 4) = *(const v4f*)(g + fo);
#endif
  }
}

// ---------------------------------------------------------------------------
// K1: fold pns1/pns2 into Mc (8 x 768):
//   Mc[f][j] = W1[j,f] + sum_k W1[k,f]*W2[j,k]   (f<7)
//   Mc[7][j] = W2[j,768]                          (flag column)
// so nde[b,v,n,j] = sum_{f=0..7} F[b,v,n,f] * Mc[f][j]
// ---------------------------------------------------------------------------
__global__ void k_mc(const float* __restrict__ w1, const float* __restrict__ w2,
                     float* __restrict__ mc) {
  int j = blockIdx.x * blockDim.x + threadIdx.x;
  if (j >= 768) return;
  float acc[7];
#pragma unroll
  for (int f = 0; f < 7; ++f) acc[f] = w1[j * 7 + f];
  const float* w2row = w2 + (size_t)j * 769;
  for (int k = 0; k < 768; ++k) {
    float w = w2row[k];
#pragma unroll
    for (int f = 0; f < 7; ++f) acc[f] = fmaf(w1[k * 7 + f], w, acc[f]);
  }
#pragma unroll
  for (int f = 0; f < 7; ++f) mc[f * 768 + j] = acc[f];
  mc[7 * 768 + j] = w2row[768];
}

// ---------------------------------------------------------------------------
// K2: query[b,v,:] = fixed_context[b] + concat(prev, veh) @ pcsᵀ
// ---------------------------------------------------------------------------
__global__ void k_query(const float* __restrict__ fixed_ctx,
                        const float* __restrict__ prev,
                        const float* __restrict__ veh,
                        const float* __restrict__ pcs,
                        float* __restrict__ query) {
  int bv = blockIdx.x;
  int b = bv / VV;
  int d = threadIdx.x;
  float acc = fixed_ctx[b * DD + d];
  const float* pr = prev + (size_t)bv * DD;
  const float* wrow = pcs + (size_t)d * (DD + 8);
  for (int e = 0; e < DD; ++e) acc = fmaf(pr[e], wrow[e], acc);
  const float* vh = veh + bv * 8;
#pragma unroll
  for (int e = 0; e < 8; ++e) acc = fmaf(vh[e], wrow[DD + e], acc);
  query[(size_t)bv * DD + d] = acc;
}

// ---------------------------------------------------------------------------
// K3: fused flash-style attention partials per (h,b,v).
// 128 threads = 4 waves; wave w owns head h = 4*(blockIdx&1)+w; block owns (b,v).
// dK/dV deltas produced with WMMA f32 16x16x4 (K=8 -> 2 k-steps); the K/V
// static tiles stream through a double-buffered LDS stage fed by
// GLOBAL_LOAD_ASYNC_TO_LDS_B128 (ASYNCcnt), prefetching tile t+1 during tile t.
// Outputs per (h,b,v): running max m, sum-exp s, unnormalized u[32].
// ---------------------------------------------------------------------------
__global__ void __launch_bounds__(128)
k_attn(const float* __restrict__ ndf, const float* __restrict__ Kst,
       const float* __restrict__ Vst, const unsigned char* __restrict__ mask,
       const float* __restrict__ mc, const float* __restrict__ query,
       float* __restrict__ u_ws, float* __restrict__ m_ws,
       float* __restrict__ s_ws) {
  __shared__ float s_compat[4][NPAD];
  __shared__ float s_tile[4][16][KSZ];
  __shared__ __align__(16) float s_stage[4][2][16][KSZ];  // K (pass1) / V (pass2)
  __shared__ float s_q[4][KSZ];
  __shared__ float s_mc[4][8][64];

  const int bv = blockIdx.x >> 1;
  const int b = bv / VV;
  const int v = bv % VV;
  const int hg = blockIdx.x & 1;
  const int w = threadIdx.x >> 5;
  const int lane = threadIdx.x & 31;
  const int h = hg * 4 + w;

  // Per-wave loads: gQ slice + Mc window (cols [h*32,+32) dV, [256+h*32,+32) dK)
  s_q[w][lane] = query[(size_t)bv * DD + h * KSZ + lane];
  for (int i = lane; i < 8 * 64; i += 32) {
    int f = i >> 6, lc = i & 63;
    int gcol = (lc < 32) ? (h * KSZ + lc) : (256 + h * KSZ + (lc - 32));
    s_mc[w][f][lc] = mc[f * 768 + gcol];
  }
  __builtin_amdgcn_wave_barrier();

  const int half = lane >> 4;   // 0/1 -> K pair select + row half of D tile
  const int r16 = lane & 15;
  const int kb = half * 2;

  const float* Fbase = ndf + (size_t)bv * NN * FNODE;
  const float* Kpl = Kst + (size_t)(h * BB + b) * NN * KSZ;  // K plane (h,b)
  const float* Vpl = Vst + (size_t)(h * BB + b) * NN * KSZ;  // V plane (h,b)
  const float rks = 0.17677669529663687f;  // 1/sqrt(32)
  float lmax = -3.0e38f;

  // ---- pass 1: compat + row max --------------------------------------
  stage_tile(Kpl, &s_stage[w][0][0][0], lane, (long)NN * KSZ - 4);
  for (int t = 0; t < NTILES; ++t) {
    const int n0 = t * 16;
    const int buf = t & 1;
    if (t + 1 < NTILES) {
      stage_tile(Kpl + (size_t)(n0 + 16) * KSZ, &s_stage[w][buf ^ 1][0][0],
                 lane, (long)(NN - (n0 + 16)) * KSZ - 4);
      ASYNC_WAIT_PREV();
    } else {
      ASYNC_WAIT_ALL();
    }
    const int nodeA = n0 + r16;
    const bool inA = nodeA < NN;
    const float* Fp = Fbase + (size_t)nodeA * FNODE;
    v2f a0, a1;
    a0.x = inA ? Fp[kb] : 0.f;     a0.y = inA ? Fp[kb + 1] : 0.f;
    a1.x = inA ? Fp[4 + kb] : 0.f; a1.y = inA ? Fp[5 + kb] : 0.f;
#pragma unroll
    for (int ct = 0; ct < 2; ++ct) {
      int lc = 32 + ct * 16 + r16;  // dK window of Mc
      v2f b0, b1;
      b0.x = s_mc[w][kb][lc];     b0.y = s_mc[w][kb + 1][lc];
      b1.x = s_mc[w][4 + kb][lc]; b1.y = s_mc[w][5 + kb][lc];
      v8f c = {};
      c = wmma4(a0, b0, c);
      c = wmma4(a1, b1, c);
#pragma unroll
      for (int r = 0; r < 8; ++r)
        s_tile[w][r + 8 * half][ct * 16 + r16] = c[r];
    }
    __builtin_amdgcn_wave_barrier();
    if (lane < 16) {
      int node = n0 + lane;
      float cval = -1.0e30f;
      if (node < NN) {
        float acc = 0.f;
#pragma unroll
        for (int k = 0; k < KSZ; ++k)
          acc = fmaf(s_q[w][k], s_stage[w][buf][lane][k] + s_tile[w][lane][k],
                     acc);
        cval = mask[(size_t)bv * NN + node] ? acc * rks : -1.0e30f;
      }
      s_compat[w][n0 + lane] = cval;
      lmax = fmaxf(lmax, cval);
    }
    __builtin_amdgcn_wave_barrier();
  }
#pragma unroll
  for (int off = 16; off >= 1; off >>= 1)
    lmax = fmaxf(lmax, __shfl_xor(lmax, off, 32));
  const float m = lmax;

  // sum-exp
  float sacc = 0.f;
  for (int i = lane; i < NPAD; i += 32) sacc += __expf(s_compat[w][i] - m);
#pragma unroll
  for (int off = 16; off >= 1; off >>= 1) sacc += __shfl_xor(sacc, off, 32);

  // ---- pass 2: u[k] = sum_n exp(c-m) * (Vst + dV) --------------------
  float u = 0.f;
  stage_tile(Vpl, &s_stage[w][0][0][0], lane, (long)NN * KSZ - 4);
  for (int t = 0; t < NTILES; ++t) {
    const int n0 = t * 16;
    const int buf = t & 1;
    if (t + 1 < NTILES) {
      stage_tile(Vpl + (size_t)(n0 + 16) * KSZ, &s_stage[w][buf ^ 1][0][0],
                 lane, (long)(NN - (n0 + 16)) * KSZ - 4);
      ASYNC_WAIT_PREV();
    } else {
      ASYNC_WAIT_ALL();
    }
    const int nodeA = n0 + r16;
    const bool inA = nodeA < NN;
    const float* Fp = Fbase + (size_t)nodeA * FNODE;
    v2f a0, a1;
    a0.x = inA ? Fp[kb] : 0.f;     a0.y = inA ? Fp[kb + 1] : 0.f;
    a1.x = inA ? Fp[4 + kb] : 0.f; a1.y = inA ? Fp[5 + kb] : 0.f;
#pragma unroll
    for (int ct = 0; ct < 2; ++ct) {
      int lc = ct * 16 + r16;  // dV window of Mc
      v2f b0, b1;
      b0.x = s_mc[w][kb][lc];     b0.y = s_mc[w][kb + 1][lc];
      b1.x = s_mc[w][4 + kb][lc]; b1.y = s_mc[w][5 + kb][lc];
      v8f c = {};
      c = wmma4(a0, b0, c);
      c = wmma4(a1, b1, c);
#pragma unroll
      for (int r = 0; r < 8; ++r)
        s_tile[w][r + 8 * half][ct * 16 + r16] = c[r];
    }
    __builtin_amdgcn_wave_barrier();
#pragma unroll 4
    for (int r = 0; r < 16; ++r) {
      int node = n0 + r;
      float wgt = (node < NN) ? __expf(s_compat[w][node] - m) : 0.f;
      u = fmaf(wgt, s_stage[w][buf][r][lane] + s_tile[w][r][lane], u);
    }
    __builtin_amdgcn_wave_barrier();
  }
  const int hbv = (h * BB + b) * VV + v;
  u_ws[(size_t)hbv * KSZ + lane] = u;
  if (lane == 0) { m_ws[hbv] = m; s_ws[hbv] = sacc; }
}

// ---------------------------------------------------------------------------
// K4: merge softmax partials across vehicles per (h,b); emit concated heads.
// ---------------------------------------------------------------------------
__global__ void k_combine(const float* __restrict__ u_ws,
                          const float* __restrict__ m_ws,
                          const float* __restrict__ s_ws,
                          float* __restrict__ conc) {
  __shared__ float sm[VV], ss[VV];
  const int hb = blockIdx.x;     // h*BB + b
  const int h = hb >> 3, b = hb & 7;
  const int tid = threadIdx.x;
  const int v = tid >> 5, k = tid & 31;
  if (tid < VV) { sm[tid] = m_ws[hb * VV + tid]; ss[tid] = s_ws[hb * VV + tid]; }
  __syncthreads();
  float M = sm[0];
#pragma unroll
  for (int i = 1; i < VV; ++i) M = fmaxf(M, sm[i]);
  float Z = 0.f;
#pragma unroll
  for (int i = 0; i < VV; ++i) Z += ss[i] * __expf(sm[i] - M);
  const float scale = __expf(sm[v] - M) / Z;
  const float hval = u_ws[(size_t)(hb * VV + v) * KSZ + k] * scale;
  conc[(size_t)(b * VV + v) * DD + h * KSZ + k] = hval;
}

// ---------------------------------------------------------------------------
// K5: final_Q = concated @ po_weightᵀ
// ---------------------------------------------------------------------------
__global__ void k_finalq(const float* __restrict__ conc,
                         const float* __restrict__ po, float* __restrict__ fq) {
  const int bv = blockIdx.x, d = threadIdx.x;
  const float* cr = conc + (size_t)bv * DD;
  const float* wr = po + (size_t)d * DD;
  float acc = 0.f;
  for (int e = 0; e < DD; ++e) acc = fmaf(cr[e], wr[e], acc);
  fq[(size_t)bv * DD + d] = acc;
}

// ---------------------------------------------------------------------------
// K6: logits[b,v,n] = 10*tanh( fQ · (lKst[n] + dlK[n]) / 16 ) + logmask
// dlK delta (nde columns 512..767) built per 16-node tile with WMMA,
// dot reduced across the 16-lane D-tile columns with shfl_xor.
// ---------------------------------------------------------------------------
__global__ void __launch_bounds__(256)
k_logits(const float* __restrict__ ndf, const float* __restrict__ lKst,
         const unsigned char* __restrict__ mask, const float* __restrict__ mc,
         const float* __restrict__ fq, float* __restrict__ logits) {
  __shared__ float s_mcl[8][256];
  __shared__ float s_fq[256];
  const int bv = blockIdx.x, b = bv / VV;
  const int tid = threadIdx.x;
  for (int i = tid; i < 8 * 256; i += 256)
    s_mcl[i >> 8][i & 255] = mc[(i >> 8) * 768 + 512 + (i & 255)];
  s_fq[tid] = fq[(size_t)bv * DD + tid];
  __syncthreads();

  const int w = tid >> 5, lane = tid & 31;
  const int half = lane >> 4, r16 = lane & 15, kb = half * 2;
  const float* Fbase = ndf + (size_t)bv * NN * FNODE;

  for (int t = w; t < NTILES; t += 8) {
    const int n0 = t * 16;
    const int nodeA = n0 + r16;
    const bool inA = nodeA < NN;
    const float* Fp = Fbase + (size_t)nodeA * FNODE;
    v2f a0, a1;
    a0.x = inA ? Fp[kb] : 0.f;     a0.y = inA ? Fp[kb + 1] : 0.f;
    a1.x = inA ? Fp[4 + kb] : 0.f; a1.y = inA ? Fp[5 + kb] : 0.f;
    float acc[8] = {0.f, 0.f, 0.f, 0.f, 0.f, 0.f, 0.f, 0.f};
    for (int cc = 0; cc < 16; ++cc) {
      const int jc = cc * 16 + r16;  // column within D
      v2f b0, b1;
      b0.x = s_mcl[kb][jc];     b0.y = s_mcl[kb + 1][jc];
      b1.x = s_mcl[4 + kb][jc]; b1.y = s_mcl[5 + kb][jc];
      v8f c = {};
      c = wmma4(a0, b0, c);
      c = wmma4(a1, b1, c);
#pragma unroll
      for (int r = 0; r < 8; ++r) {
        int node = n0 + r + 8 * half;
        if (node < NN)
          acc[r] = fmaf(s_fq[jc],
                        lKst[((size_t)b * NN + node) * DD + jc] + c[r], acc[r]);
      }
    }
#pragma unroll
    for (int r = 0; r < 8; ++r)
#pragma unroll
      for (int off = 1; off <= 8; off <<= 1)
        acc[r] += __shfl_xor(acc[r], off, 32);
    if (r16 == 0) {
#pragma unroll
      for (int r = 0; r < 8; ++r) {
        int node = n0 + r + 8 * half;
        if (node < NN) {
          float l = tanhf(acc[r] * 0.0625f) * 10.0f;
          logits[(size_t)bv * NN + node] =
              mask[(size_t)bv * NN + node] ? l : -1.0e30f;
        }
      }
    }
  }
}

// ---------------------------------------------------------------------------
// K7: per-batch softmax over V*N, argmax, logprob; stash prob*logprob.
// ---------------------------------------------------------------------------
__global__ void k_select(const float* __restrict__ logits,
                         float* __restrict__ out, float* __restrict__ pl) {
  __shared__ float smax[256];
  __shared__ int sarg[256];
  __shared__ float ssum[256];
  const int b = blockIdx.x, tid = threadIdx.x;
  const float* lb = logits + (size_t)b * VV * NN;
  float mx = -3.0e38f;
  int arg = 0;
  for (int i = tid; i < VV * NN; i += 256) {
    float vl = lb[i];
    if (vl > mx) { mx = vl; arg = i; }
  }
  smax[tid] = mx; sarg[tid] = arg;
  __syncthreads();
  for (int s = 128; s > 0; s >>= 1) {
    if (tid < s && smax[tid + s] > smax[tid]) {
      smax[tid] = smax[tid + s];
      sarg[tid] = sarg[tid + s];
    }
    __syncthreads();
  }
  const float M = smax[0];
  const int op = sarg[0];
  float z = 0.f;
  for (int i = tid; i < VV * NN; i += 256) z += __expf(lb[i] - M);
  ssum[tid] = z;
  __syncthreads();
  for (int s = 128; s > 0; s >>= 1) {
    if (tid < s) ssum[tid] += ssum[tid + s];
    __syncthreads();
  }
  if (tid == 0) {
    const float Z = ssum[0];
    const float logprob = -__logf(Z);       // max prob = exp(M-M)/Z = 1/Z
    out[b] = (float)(op / NN);              // selected_vecs
    out[8 + b] = (float)(op % NN);          // selected_nodes
    out[16 + b] = logprob;                  // logprob
    pl[b] = (1.0f / Z) * logprob;           // prob*logprob for entropy
  }
}

__global__ void k_entropy(const float* __restrict__ pl, float* __restrict__ out) {
  if (threadIdx.x == 0) {
    float e = 0.f;
#pragma unroll
    for (int b = 0; b < BB; ++b) e += pl[b];
    out[24] = -e;
  }
}

// ---------------------------------------------------------------------------
extern "C" void kernel_launch(void* const* d_in, const int* in_sizes, int n_in,
                              void* d_out, int out_size, void* d_ws,
                              size_t ws_size, hipStream_t stream) {
  // d_in order follows setup_inputs(): node_embeddings is unused by reference.
  const float* fixed_ctx = (const float*)d_in[1];
  const float* prev = (const float*)d_in[2];
  const float* ndf = (const float*)d_in[3];
  const float* veh = (const float*)d_in[4];
  const float* Vst = (const float*)d_in[5];
  const float* Kst = (const float*)d_in[6];
  const float* lKst = (const float*)d_in[7];
  const unsigned char* mask = (const unsigned char*)d_in[8];  // jax bool -> 1B
  const float* pcs = (const float*)d_in[9];
  const float* w1 = (const float*)d_in[10];
  const float* w2 = (const float*)d_in[11];
  const float* po = (const float*)d_in[12];

  float* ws = (float*)d_ws;
  float* mc = ws;                  //  8*768      = 6144
  float* query = ws + 6144;        //  64*256     = 16384
  float* u_ws = ws + 22528;        //  512*32     = 16384
  float* m_ws = ws + 38912;        //  512
  float* s_ws = ws + 39424;        //  512
  float* conc = ws + 39936;        //  64*256
  float* fqb = ws + 56320;         //  64*256
  float* logits = ws + 72704;      //  64*1000
  float* pl = ws + 136704;         //  8
  float* out = (float*)d_out;      //  25 floats: vecs(8)|nodes(8)|logprob(8)|entropy(1)

  k_mc<<<6, 128, 0, stream>>>(w1, w2, mc);
  k_query<<<BB * VV, 256, 0, stream>>>(fixed_ctx, prev, veh, pcs, query);
  k_attn<<<BB * VV * 2, 128, 0, stream>>>(ndf, Kst, Vst, mask, mc, query, u_ws,
                                          m_ws, s_ws);
  k_combine<<<HH * BB, 256, 0, stream>>>(u_ws, m_ws, s_ws, conc);
  k_finalq<<<BB * VV, 256, 0, stream>>>(conc, po, fqb);
  k_logits<<<BB * VV, 256, 0, stream>>>(ndf, lKst, mask, mc, fqb, logits);
  k_select<<<BB, 256, 0, stream>>>(logits, out, pl);
  k_entropy<<<1, 32, 0, stream>>>(pl, out);
}